// GCN_31370441130271
// MI455X (gfx1250) — compile-verified
//
#include <hip/hip_runtime.h>
#include <hip/hip_bf16.h>
#include <math.h>

typedef __attribute__((ext_vector_type(2))) float v2f;
typedef __attribute__((ext_vector_type(8))) float v8f;

// ---------------------------------------------------------------------------
// Zero-fill (float4 granularity; all sizes here are multiples of 4 floats)
// ---------------------------------------------------------------------------
__global__ void gcn_zero_f4(float4* __restrict__ p, long n4) {
    long i = (long)blockIdx.x * blockDim.x + threadIdx.x;
    if (i < n4) p[i] = make_float4(0.f, 0.f, 0.f, 0.f);
}

// ---------------------------------------------------------------------------
// GEMM1: h0 = x @ W1   (x: [N,256], W1: [256,128], h0: [N,128])
// One wave per 16x16 output tile, fp32 WMMA 16x16x4, K-loop of 64 steps.
// Block = 256 threads = 8 waves -> the 8 column tiles of one 16-row stripe.
// ---------------------------------------------------------------------------
__global__ void gcn_gemm1_wmma(const float* __restrict__ x,
                               const float* __restrict__ W1,
                               float* __restrict__ h0) {
    const int wave = threadIdx.x >> 5;
    const int lane = threadIdx.x & 31;
    const int row0 = blockIdx.x * 16;
    const int col0 = wave * 16;        // 8 waves cover 128 cols

    const int mn = lane & 15;          // A: row within tile, B: col within tile
    const int kg = lane >> 4;          // K-pair group (0 or 1)

    const float* __restrict__ arow = x  + (size_t)(row0 + mn) * 256;
    const float* __restrict__ bcol = W1 + (col0 + mn);

    v8f c = {0.f, 0.f, 0.f, 0.f, 0.f, 0.f, 0.f, 0.f};

    #pragma unroll 4
    for (int k = 0; k < 256; k += 4) {
        const int ka = k + 2 * kg;
        v2f a, b;
        a.x = arow[ka];
        a.y = arow[ka + 1];
        b.x = bcol[(size_t)ka * 128];
        b.y = bcol[(size_t)(ka + 1) * 128];
        c = __builtin_amdgcn_wmma_f32_16x16x4_f32(
                /*neg_a=*/false, a, /*neg_b=*/false, b,
                /*c_mod=*/(short)0, c, /*reuse_a=*/false, /*reuse_b=*/false);
    }

    // C/D layout: VGPR r -> M = r + 8*(lane>=16), N = lane&15
    const int mb = (lane < 16) ? 0 : 8;
    #pragma unroll
    for (int r = 0; r < 8; ++r) {
        h0[(size_t)(row0 + mb + r) * 128 + col0 + mn] = c[r];
    }
}

// ---------------------------------------------------------------------------
// SpMM, F=128: one wave per edge; lane handles a contiguous float4.
// dst must be pre-zeroed; FP32 hardware atomics (global_atomic_add_f32).
// ---------------------------------------------------------------------------
__global__ void gcn_spmm128(const int* __restrict__ rows,
                            const int* __restrict__ cols,
                            const float* __restrict__ w,
                            const float* __restrict__ src,
                            float* __restrict__ dst, int E) {
    const int e = blockIdx.x * (blockDim.x >> 5) + (threadIdx.x >> 5);
    if (e >= E) return;
    const int lane = threadIdx.x & 31;
    const int r = rows[e];
    const int cidx = cols[e];
    const float we = w[e];
    const float4 v = *(const float4*)(src + (size_t)cidx * 128 + lane * 4);
    float* d = dst + (size_t)r * 128 + lane * 4;
    unsafeAtomicAdd(d + 0, v.x * we);
    unsafeAtomicAdd(d + 1, v.y * we);
    unsafeAtomicAdd(d + 2, v.z * we);
    unsafeAtomicAdd(d + 3, v.w * we);
}

// ---------------------------------------------------------------------------
// GEMM2: out0 = relu(h + b1) @ W2   (h: [N,128], W2: [128,40], out0: [N,40])
// Bias + ReLU fused into A-fragment load. Column dim padded 40->48 (3 tiles),
// out-of-range columns read/store-guarded. grid = (N/16, 3), block = 1 wave.
// ---------------------------------------------------------------------------
__global__ void gcn_gemm2_wmma(const float* __restrict__ h,
                               const float* __restrict__ b1,
                               const float* __restrict__ W2,
                               float* __restrict__ out0) {
    const int lane = threadIdx.x;      // 0..31
    const int row0 = blockIdx.x * 16;
    const int col0 = blockIdx.y * 16;

    const int mn = lane & 15;
    const int kg = lane >> 4;
    const int cb = col0 + mn;
    const bool cvalid = (cb < 40);

    const float* __restrict__ arow = h + (size_t)(row0 + mn) * 128;

    v8f c = {0.f, 0.f, 0.f, 0.f, 0.f, 0.f, 0.f, 0.f};

    #pragma unroll 4
    for (int k = 0; k < 128; k += 4) {
        const int ka = k + 2 * kg;
        v2f a, b;
        a.x = fmaxf(arow[ka]     + b1[ka],     0.f);
        a.y = fmaxf(arow[ka + 1] + b1[ka + 1], 0.f);
        b.x = cvalid ? W2[(size_t)ka * 40 + cb]       : 0.f;
        b.y = cvalid ? W2[(size_t)(ka + 1) * 40 + cb] : 0.f;
        c = __builtin_amdgcn_wmma_f32_16x16x4_f32(
                false, a, false, b, (short)0, c, false, false);
    }

    if (cvalid) {
        const int mb = (lane < 16) ? 0 : 8;
        #pragma unroll
        for (int r = 0; r < 8; ++r) {
            out0[(size_t)(row0 + mb + r) * 40 + cb] = c[r];
        }
    }
}

// ---------------------------------------------------------------------------
// SpMM, F=40: one wave per edge; lanes cover f=lane (0..31) and f=32+lane (<8).
// ---------------------------------------------------------------------------
__global__ void gcn_spmm40(const int* __restrict__ rows,
                           const int* __restrict__ cols,
                           const float* __restrict__ w,
                           const float* __restrict__ src,
                           float* __restrict__ dst, int E) {
    const int e = blockIdx.x * (blockDim.x >> 5) + (threadIdx.x >> 5);
    if (e >= E) return;
    const int lane = threadIdx.x & 31;
    const int r = rows[e];
    const int cidx = cols[e];
    const float we = w[e];
    const float* s = src + (size_t)cidx * 40;
    float* d = dst + (size_t)r * 40;
    unsafeAtomicAdd(d + lane, s[lane] * we);
    if (lane < 8) {
        unsafeAtomicAdd(d + 32 + lane, s[32 + lane] * we);
    }
}

// ---------------------------------------------------------------------------
// Log-softmax over 40 classes, fused +b2, in-place on d_out. One wave per row.
// ---------------------------------------------------------------------------
__global__ void gcn_logsoftmax40(float* __restrict__ out,
                                 const float* __restrict__ b2, int N) {
    const int r = blockIdx.x * (blockDim.x >> 5) + (threadIdx.x >> 5);
    if (r >= N) return;
    const int lane = threadIdx.x & 31;
    float* row = out + (size_t)r * 40;

    const float v0 = row[lane] + b2[lane];
    const float v1 = (lane < 8) ? (row[32 + lane] + b2[32 + lane]) : -INFINITY;

    float mx = fmaxf(v0, v1);
    #pragma unroll
    for (int o = 16; o > 0; o >>= 1) mx = fmaxf(mx, __shfl_xor(mx, o, 32));

    float s = expf(v0 - mx) + ((lane < 8) ? expf(v1 - mx) : 0.f);
    #pragma unroll
    for (int o = 16; o > 0; o >>= 1) s += __shfl_xor(s, o, 32);

    const float lse = mx + logf(s);
    row[lane] = v0 - lse;
    if (lane < 8) row[32 + lane] = v1 - lse;
}

// ---------------------------------------------------------------------------
// Launcher
// Inputs: 0:x [N,256]  1:edge_row [E]  2:edge_col [E]  3:edge_w [E]
//         4:W1 [256,128]  5:b1 [128]  6:W2 [128,40]  7:b2 [40]
// Output: [N,40] log-softmax
// ---------------------------------------------------------------------------
extern "C" void kernel_launch(void* const* d_in, const int* in_sizes, int n_in,
                              void* d_out, int out_size, void* d_ws, size_t ws_size,
                              hipStream_t stream) {
    const float* x        = (const float*)d_in[0];
    const int*   edge_row = (const int*)  d_in[1];
    const int*   edge_col = (const int*)  d_in[2];
    const float* edge_w   = (const float*)d_in[3];
    const float* W1       = (const float*)d_in[4];
    const float* b1       = (const float*)d_in[5];
    const float* W2       = (const float*)d_in[6];
    const float* b2       = (const float*)d_in[7];

    const int N = in_sizes[0] / 256;   // 100000
    const int E = in_sizes[1];         // 1600000

    float* h0  = (float*)d_ws;                       // [N,128]; reused as out0 [N,40]
    float* h   = h0 + (size_t)N * 128;               // [N,128]
    float* out = (float*)d_out;                      // [N,40]

    const int rowTiles = N / 16;                     // 6250

    // 1) zero h (SpMM1 target)
    {
        long n4 = ((long)N * 128) / 4;
        gcn_zero_f4<<<(int)((n4 + 255) / 256), 256, 0, stream>>>((float4*)h, n4);
    }
    // 2) GEMM1: h0 = x @ W1
    gcn_gemm1_wmma<<<rowTiles, 256, 0, stream>>>(x, W1, h0);
    // 3) SpMM1: h += scatter(w * h0[col])
    {
        int edgesPerBlock = 256 / 32;
        gcn_spmm128<<<(E + edgesPerBlock - 1) / edgesPerBlock, 256, 0, stream>>>(
            edge_row, edge_col, edge_w, h0, h, E);
    }
    // 4) GEMM2: out0 = relu(h + b1) @ W2   (out0 reuses h0's storage)
    {
        dim3 grid(rowTiles, 3);
        gcn_gemm2_wmma<<<grid, 32, 0, stream>>>(h, b1, W2, h0);
    }
    // 5) zero d_out (SpMM2 target)
    {
        long n4 = ((long)N * 40) / 4;
        gcn_zero_f4<<<(int)((n4 + 255) / 256), 256, 0, stream>>>((float4*)out, n4);
    }
    // 6) SpMM2: out += scatter(w * out0[col])
    {
        int edgesPerBlock = 256 / 32;
        gcn_spmm40<<<(E + edgesPerBlock - 1) / edgesPerBlock, 256, 0, stream>>>(
            edge_row, edge_col, edge_w, h0, out, E);
    }
    // 7) log_softmax(out + b2), in place
    {
        int rowsPerBlock = 256 / 32;
        gcn_logsoftmax40<<<(N + rowsPerBlock - 1) / rowsPerBlock, 256, 0, stream>>>(
            out, b2, N);
    }
}